// Gemma3Attention_41016937677366
// MI455X (gfx1250) — compile-verified
//
#include <hip/hip_runtime.h>
#include <hip/hip_bf16.h>

typedef __attribute__((ext_vector_type(2))) float v2f;
typedef __attribute__((ext_vector_type(8))) float v8f;

// D = A(16x4 f32) * B(4x16 f32) + C(16x16 f32)  -- CDNA5 fp32 WMMA
static __device__ inline v8f wmma4(v2f a, v2f b, v8f c) {
    return __builtin_amdgcn_wmma_f32_16x16x4_f32(
        false, a, false, b, (short)0, c, false, false);
}

#define T_LEN 2048
#define HID   2048
#define NH    16
#define NHK   8
#define HD    128
#define NQKV  4096   // (H + 2*HK) * D

// ---------------------------------------------------------------------------
// Row-major fp32 GEMM: C[M,N] = A[M,K] @ B[K,N] via WMMA f32 16x16x4.
// Block 256 thr (8 waves), tile 128x128, BK=16, double-buffered LDS.
// Wave tile 32x64 -> 8 accumulators; 8 WMMAs per 6 ds_load_b64.
// B kept transposed in LDS so every fragment is one aligned b64 load.
// ---------------------------------------------------------------------------
#define BM 128
#define BN 128
#define BK 16
#define AST 20   // As row stride (16B-aligned rows, conflict-free reads)
#define BST 18   // BsT row stride (8B-aligned b64 reads, conflict-free)

__global__ __launch_bounds__(256)
void gemm_f32_wmma(const float* __restrict__ A, const float* __restrict__ B,
                   float* __restrict__ C, int M, int N, int K) {
    __shared__ float As[2][BM][AST];
    __shared__ float BsT[2][BN][BST];

    const int tid  = threadIdx.x;
    const int wave = tid >> 5;
    const int lane = tid & 31;
    const int lo   = lane & 15;
    const int hlf  = lane >> 4;

    const int tm = blockIdx.y * BM;
    const int tn = blockIdx.x * BN;
    const int wm = (wave >> 1) * 32;   // 0,32,64,96
    const int wn = (wave & 1) * 64;    // 0,64

    // staging coordinates (A: 128x16 = 512 float4; B: 16x128 = 512 float4)
    const int ar = tid >> 2;           // A row 0..63 (+64 second chunk)
    const int ac = (tid & 3) * 4;      // A col
    const int br = tid >> 5;           // B row 0..7 (+8 second chunk)
    const int bc = (tid & 31) * 4;     // B col

    const v8f vzero = {0.f,0.f,0.f,0.f,0.f,0.f,0.f,0.f};
    v8f acc[2][4];
    #pragma unroll
    for (int mi = 0; mi < 2; ++mi)
        #pragma unroll
        for (int j = 0; j < 4; ++j) acc[mi][j] = vzero;

    float4 aR0, aR1, bR0, bR1;

    // prologue: stage tile 0 into buffer 0
    aR0 = *(const float4*)(A + (size_t)(tm + ar)      * K + ac);
    aR1 = *(const float4*)(A + (size_t)(tm + ar + 64) * K + ac);
    bR0 = *(const float4*)(B + (size_t)(br)     * N + tn + bc);
    bR1 = *(const float4*)(B + (size_t)(br + 8) * N + tn + bc);
    *(float4*)&As[0][ar][ac]      = aR0;
    *(float4*)&As[0][ar + 64][ac] = aR1;
    BsT[0][bc + 0][br] = bR0.x; BsT[0][bc + 1][br] = bR0.y;
    BsT[0][bc + 2][br] = bR0.z; BsT[0][bc + 3][br] = bR0.w;
    BsT[0][bc + 0][br + 8] = bR1.x; BsT[0][bc + 1][br + 8] = bR1.y;
    BsT[0][bc + 2][br + 8] = bR1.z; BsT[0][bc + 3][br + 8] = bR1.w;
    __syncthreads();

    const int nk = K / BK;
    for (int kt = 0; kt < nk; ++kt) {
        const int cur = kt & 1;
        if (kt + 1 < nk) {  // prefetch next tile into registers during compute
            const int k0 = (kt + 1) * BK;
            aR0 = *(const float4*)(A + (size_t)(tm + ar)      * K + k0 + ac);
            aR1 = *(const float4*)(A + (size_t)(tm + ar + 64) * K + k0 + ac);
            bR0 = *(const float4*)(B + (size_t)(k0 + br)     * N + tn + bc);
            bR1 = *(const float4*)(B + (size_t)(k0 + br + 8) * N + tn + bc);
        }
        #pragma unroll
        for (int kk = 0; kk < BK; kk += 4) {
            const int kd = kk + hlf * 2;
            v2f a0 = { As[cur][wm + lo][kd],      As[cur][wm + lo][kd + 1] };
            v2f a1 = { As[cur][wm + 16 + lo][kd], As[cur][wm + 16 + lo][kd + 1] };
            #pragma unroll
            for (int j = 0; j < 4; ++j) {
                v2f b = { BsT[cur][wn + j * 16 + lo][kd],
                          BsT[cur][wn + j * 16 + lo][kd + 1] };
                acc[0][j] = wmma4(a0, b, acc[0][j]);
                acc[1][j] = wmma4(a1, b, acc[1][j]);
            }
        }
        if (kt + 1 < nk) {
            const int nxt = cur ^ 1;
            *(float4*)&As[nxt][ar][ac]      = aR0;
            *(float4*)&As[nxt][ar + 64][ac] = aR1;
            BsT[nxt][bc + 0][br] = bR0.x; BsT[nxt][bc + 1][br] = bR0.y;
            BsT[nxt][bc + 2][br] = bR0.z; BsT[nxt][bc + 3][br] = bR0.w;
            BsT[nxt][bc + 0][br + 8] = bR1.x; BsT[nxt][bc + 1][br + 8] = bR1.y;
            BsT[nxt][bc + 2][br + 8] = bR1.z; BsT[nxt][bc + 3][br + 8] = bR1.w;
        }
        __syncthreads();
    }

    #pragma unroll
    for (int mi = 0; mi < 2; ++mi) {
        const int row0 = tm + wm + mi * 16 + hlf * 8;
        #pragma unroll
        for (int r = 0; r < 8; ++r)
            #pragma unroll
            for (int j = 0; j < 4; ++j)
                C[(size_t)(row0 + r) * N + tn + wn + j * 16 + lo] = acc[mi][j][r];
    }
}

// ---------------------------------------------------------------------------
// Per-head Gemma RMSNorm + neox RoPE, in place on qkv. One wave per (t, head).
// q heads also absorb the attention scale D^-0.5.
// ---------------------------------------------------------------------------
__global__ __launch_bounds__(128)
void norm_rope(float* __restrict__ qkv, const int* __restrict__ pos,
               const float* __restrict__ qw, const float* __restrict__ kw) {
    const int gw   = blockIdx.x * 4 + (threadIdx.x >> 5);
    const int lane = threadIdx.x & 31;
    const int t = gw / (NH + NHK);
    const int h = gw % (NH + NHK);
    const bool is_q = (h < NH);

    float* vec = qkv + (size_t)t * NQKV + (is_q ? h * HD : NH * HD + (h - NH) * HD);
    const float* w = is_q ? qw : kw;

    const int i0 = lane * 2;                       // pair indices i0, i0+1 (0..63)
    float2 x1 = *(const float2*)(vec + i0);        // first half
    float2 x2 = *(const float2*)(vec + 64 + i0);   // second half

    float ss = x1.x*x1.x + x1.y*x1.y + x2.x*x2.x + x2.y*x2.y;
    #pragma unroll
    for (int off = 16; off >= 1; off >>= 1)
        ss += __shfl_xor(ss, off, 32);
    const float inv = rsqrtf(ss * (1.0f / 128.0f) + 1e-6f);

    float n1x = x1.x * inv * (1.0f + w[i0]);
    float n1y = x1.y * inv * (1.0f + w[i0 + 1]);
    float n2x = x2.x * inv * (1.0f + w[64 + i0]);
    float n2y = x2.y * inv * (1.0f + w[64 + i0 + 1]);

    const float p = (float)pos[t];
    const float lnth_over = 9.210340371976184f / 64.0f;  // ln(10000)/64
    float f0 = __expf(-(float)i0 * lnth_over);
    float f1 = __expf(-(float)(i0 + 1) * lnth_over);
    float a0 = p * f0, a1 = p * f1;
    float c0 = __cosf(a0), s0 = __sinf(a0);
    float c1 = __cosf(a1), s1 = __sinf(a1);

    const float scl = is_q ? 0.08838834764831845f : 1.0f;  // D^-0.5 folded into q
    vec[i0]          = (n1x * c0 - n2x * s0) * scl;
    vec[i0 + 1]      = (n1y * c1 - n2y * s1) * scl;
    vec[64 + i0]     = (n2x * c0 + n1x * s0) * scl;
    vec[64 + i0 + 1] = (n2y * c1 + n1y * s1) * scl;
}

// ---------------------------------------------------------------------------
// Flash attention. grid = (T/128, H). 256 thr = 8 waves; wave owns 16 q rows.
// K tile staged global->LDS with GLOBAL_LOAD_ASYNC_TO_LDS_B128 (ASYNCcnt),
// V tile staged transposed through VGPRs, P re-fragmented via LDS.
// ---------------------------------------------------------------------------
__global__ __launch_bounds__(256)
void attn(const float* __restrict__ qkv, float* __restrict__ o) {
    __shared__ float Kt[16][132];     // row-major K tile  [key][d]
    __shared__ float VtT[128][18];    // transposed V tile [d][key]
    __shared__ float sP[8][16][18];   // per-wave P tile   [row][key]

    const int h  = blockIdx.y;
    const int hk = h >> 1;               // GQA: 2 q heads per kv head
    const int qb = blockIdx.x;
    const int tid  = threadIdx.x;
    const int wave = tid >> 5;
    const int lane = tid & 31;
    const int lo   = lane & 15;
    const int hlf  = lane >> 4;

    const int m0 = qb * 128 + wave * 16;  // wave's q row base

    // preload q A-fragments (already RMSNormed + RoPEd + scaled)
    v2f qa[32];
    const float* qp = qkv + (size_t)(m0 + lo) * NQKV + h * HD;
    #pragma unroll
    for (int c = 0; c < 32; ++c) {
        float2 t2 = *(const float2*)(qp + c * 4 + hlf * 2);
        qa[c] = (v2f){ t2.x, t2.y };
    }

    const v8f vzero = {0.f,0.f,0.f,0.f,0.f,0.f,0.f,0.f};
    v8f accO[8];
    #pragma unroll
    for (int j = 0; j < 8; ++j) accO[j] = vzero;
    float mrow[8], lrow[8];
    #pragma unroll
    for (int r = 0; r < 8; ++r) { mrow[r] = -3.0e38f; lrow[r] = 0.f; }

    const float* kbase = qkv + NH * HD + hk * HD;            // k block
    const float* vbase = qkv + NH * HD + NHK * HD + hk * HD; // v block

    const int kr = tid >> 5;          // staging row 0..7 (+8)
    const int kc = (tid & 31) * 4;    // staging col

    const int ntiles = (qb + 1) * 8;  // causal: keys up to end of q block
    for (int kt = 0; kt < ntiles; ++kt) {
        const int k0 = kt * 16;

        // --- K tile: async global->LDS copy (16B per lane per op) ---
        {
            unsigned lds0 = (unsigned)(size_t)&Kt[kr][kc];
            unsigned lds1 = (unsigned)(size_t)&Kt[kr + 8][kc];
            unsigned long long g0 =
                (unsigned long long)(kbase + (size_t)(k0 + kr) * NQKV + kc);
            unsigned long long g1 =
                (unsigned long long)(kbase + (size_t)(k0 + kr + 8) * NQKV + kc);
            asm volatile("global_load_async_to_lds_b128 %0, %1, off"
                         :: "v"(lds0), "v"(g0) : "memory");
            asm volatile("global_load_async_to_lds_b128 %0, %1, off"
                         :: "v"(lds1), "v"(g1) : "memory");
        }
        // --- V tile: load and store transposed [d][key] ---
        {
            float4 v0 = *(const float4*)(vbase + (size_t)(k0 + kr) * NQKV + kc);
            float4 v1 = *(const float4*)(vbase + (size_t)(k0 + kr + 8) * NQKV + kc);
            VtT[kc + 0][kr] = v0.x; VtT[kc + 1][kr] = v0.y;
            VtT[kc + 2][kr] = v0.z; VtT[kc + 3][kr] = v0.w;
            VtT[kc + 0][kr + 8] = v1.x; VtT[kc + 1][kr + 8] = v1.y;
            VtT[kc + 2][kr + 8] = v1.z; VtT[kc + 3][kr + 8] = v1.w;
        }
        asm volatile("s_wait_asynccnt 0" ::: "memory");  // our async K copies done
        __syncthreads();

        if (k0 <= m0 + 15) {  // wave-uniform causal skip (EXEC stays full)
            // S = q @ k^T : accumulate over D=128 in K-chunks of 4
            v8f s = vzero;
            #pragma unroll
            for (int c = 0; c < 32; ++c) {
                int kd = c * 4 + hlf * 2;
                v2f b = { Kt[lo][kd], Kt[lo][kd + 1] };
                s = wmma4(qa[c], b, s);
            }

            // online softmax per row (row stats per lane, per half)
            const int col = k0 + lo;
            #pragma unroll
            for (int r = 0; r < 8; ++r) {
                const int row = m0 + r + hlf * 8;
                float sv = s[r];
                if (col > row) sv = -3.0e38f;
                float mx = sv;
                mx = fmaxf(mx, __shfl_xor(mx, 1, 32));
                mx = fmaxf(mx, __shfl_xor(mx, 2, 32));
                mx = fmaxf(mx, __shfl_xor(mx, 4, 32));
                mx = fmaxf(mx, __shfl_xor(mx, 8, 32));
                float mnew  = fmaxf(mrow[r], mx);
                float alpha = __expf(mrow[r] - mnew);
                float pv    = __expf(sv - mnew);
                float rs = pv;
                rs += __shfl_xor(rs, 1, 32);
                rs += __shfl_xor(rs, 2, 32);
                rs += __shfl_xor(rs, 4, 32);
                rs += __shfl_xor(rs, 8, 32);
                lrow[r] = lrow[r] * alpha + rs;
                mrow[r] = mnew;
                #pragma unroll
                for (int j = 0; j < 8; ++j) accO[j][r] *= alpha;
                sP[wave][r + hlf * 8][lo] = pv;   // C-layout -> plain [row][key]
            }

            // O += P @ V  (P re-fragmented from LDS; wave-local, DS in-order)
            #pragma unroll
            for (int kb = 0; kb < 16; kb += 4) {
                int kd = kb + hlf * 2;
                v2f pa = { sP[wave][lo][kd], sP[wave][lo][kd + 1] };
                #pragma unroll
                for (int j = 0; j < 8; ++j) {
                    v2f vb = { VtT[j * 16 + lo][kd], VtT[j * 16 + lo][kd + 1] };
                    accO[j] = wmma4(pa, vb, accO[j]);
                }
            }
        }
        __syncthreads();
    }

    // epilogue: divide by l, write o[row][h*128 + d]
    #pragma unroll
    for (int r = 0; r < 8; ++r) {
        const float invl = 1.0f / lrow[r];
        const int row = m0 + r + hlf * 8;
        #pragma unroll
        for (int j = 0; j < 8; ++j)
            o[(size_t)row * (NH * HD) + h * HD + j * 16 + lo] = accO[j][r] * invl;
    }
}

// ---------------------------------------------------------------------------
extern "C" void kernel_launch(void* const* d_in, const int* in_sizes, int n_in,
                              void* d_out, int out_size, void* d_ws, size_t ws_size,
                              hipStream_t stream) {
    const float* x     = (const float*)d_in[0];
    const int*   pos   = (const int*)  d_in[1];
    const float* w_qkv = (const float*)d_in[2];
    const float* w_o   = (const float*)d_in[3];
    const float* qnw   = (const float*)d_in[4];
    const float* knw   = (const float*)d_in[5];
    float* out = (float*)d_out;

    float* qkv  = (float*)d_ws;                       // [T, 4096]  32 MB
    float* obuf = qkv + (size_t)T_LEN * NQKV;         // [T, 2048]  16 MB

    // 1) qkv = x @ w_qkv
    gemm_f32_wmma<<<dim3(NQKV / BN, T_LEN / BM), 256, 0, stream>>>(
        x, w_qkv, qkv, T_LEN, NQKV, HID);
    // 2) per-head RMSNorm + RoPE (+ q scale), in place
    norm_rope<<<(T_LEN * (NH + NHK)) / 4, 128, 0, stream>>>(qkv, pos, qnw, knw);
    // 3) flash attention -> obuf
    attn<<<dim3(T_LEN / 128, NH), 256, 0, stream>>>(qkv, obuf);
    // 4) out = obuf @ w_o
    gemm_f32_wmma<<<dim3(HID / BN, T_LEN / BM), 256, 0, stream>>>(
        obuf, w_o, out, T_LEN, HID, HID);
}